// StandardChaoticCipher_78426102825404
// MI455X (gfx1250) — compile-verified
//
#include <hip/hip_runtime.h>
#include <stdint.h>

// Problem geometry (reference: B=64, C=3, H=W=256)
#define HW_N    256
#define HW_TOT  (HW_N * HW_N)   // 65536
#define TILE    2048            // chaos floats staged to LDS per block
#define WARMUP  200

// ---------------------------------------------------------------------------
// Kernel A (emitted first so the disasm snippet shows the async-LDS path):
// inverse-Arnold gather + chaos-mask add.
// grid = (HW_TOT/TILE, B), block = 256. Stages the chaos tile into LDS via
// gfx1250 GLOBAL_LOAD_ASYNC_TO_LDS_B128 (ASYNCcnt path), reused across the
// 3 channels (3x fewer global mask reads).
// ---------------------------------------------------------------------------
__global__ __launch_bounds__(256) void cipher_apply_kernel(
    const float* __restrict__ images, const float* __restrict__ chaos,
    const int* __restrict__ iters_p, const float* __restrict__ strength_p,
    float* __restrict__ out) {
  __shared__ float smem[TILE];

  const int tile = blockIdx.x;
  const int b    = blockIdx.y;
  const int tid  = threadIdx.x;

  const float* cbase = chaos + (size_t)b * HW_TOT + (size_t)tile * TILE;

  // Async-stage TILE floats (8 KB) into LDS: 2 x B128 per lane.
#pragma unroll
  for (int j = 0; j < 2; ++j) {
    int fidx = (tid + j * 256) * 4;  // float index, 16B aligned
    uint64_t gaddr = (uint64_t)(uintptr_t)(cbase + fidx);
    uint32_t laddr = (uint32_t)(uintptr_t)(&smem[fidx]);
    asm volatile("global_load_async_to_lds_b128 %0, %1, off"
                 :
                 : "v"(laddr), "v"(gaddr)
                 : "memory");
  }
#if __has_builtin(__builtin_amdgcn_s_wait_asynccnt)
  __builtin_amdgcn_s_wait_asynccnt(0);
#else
  asm volatile("s_wait_asynccnt 0" ::: "memory");
#endif
  __syncthreads();

  const int   iters    = iters_p[0];
  const float strength = strength_p[0];

#pragma unroll
  for (int c = 0; c < 3; ++c) {
    const float* __restrict__ img = images + ((size_t)b * 3 + c) * HW_TOT;
    float* __restrict__ o = out + ((size_t)b * 3 + c) * HW_TOT + (size_t)tile * TILE;
#pragma unroll
    for (int j = 0; j < 8; ++j) {
      int p  = tid + j * 256;        // 0..2047 within tile
      int hw = tile * TILE + p;      // output flat pixel
      int jx = hw & (HW_N - 1);
      int jy = hw >> 8;
      // Inverse Arnold map applied 'iters' times:
      // (x,y) <- ((2x - y) mod N, (y - x) mod N)
      int cx = jx, cy = jy;
      for (int it = 0; it < iters; ++it) {
        int nx = (2 * cx - cy) & (HW_N - 1);
        int ny = (cy - cx) & (HW_N - 1);
        cx = nx;
        cy = ny;
      }
      float src = img[(cy << 8) + cx];
      float v   = fmaf(smem[p], strength, src);
      o[p] = __builtin_amdgcn_fractf(v);
    }
  }
}

// ---------------------------------------------------------------------------
// One chaotic-map step for a 5-element state held in registers.
// sin(pi*x) == v_sin_f32(0.5*x) since HW trans unit computes sin(2*pi*s).
// '% 1.0' on any sign == fract (x - floor(x)) -> v_fract_f32.
// The 10 trans ops per step are independent (issue back-to-back, quarter-rate
// trans pipe co-executes with the full-rate fma/fract stream); the outermost
// fma consumes the trans result last, keeping the serial chain minimal.
// ---------------------------------------------------------------------------
#define CHAOS_STEP()                                                          \
  do {                                                                        \
    float s0 = __builtin_amdgcn_sinf(0.5f * x0);                              \
    float s1 = __builtin_amdgcn_sinf(0.5f * x1);                              \
    float s2 = __builtin_amdgcn_sinf(0.5f * x2);                              \
    float s3 = __builtin_amdgcn_sinf(0.5f * x3);                              \
    float s4 = __builtin_amdgcn_sinf(0.5f * x4);                              \
    float c0 = __builtin_amdgcn_cosf(0.5f * x0);                              \
    float c1 = __builtin_amdgcn_cosf(0.5f * x1);                              \
    float c2 = __builtin_amdgcn_cosf(0.5f * x2);                              \
    float c3 = __builtin_amdgcn_cosf(0.5f * x3);                              \
    float c4 = __builtin_amdgcn_cosf(0.5f * x4);                              \
    float n0 = fmaf(mu, s0, fmaf(alpha, x1 - x4, fmaf(beta, c2, 0.1f * x0))); \
    float n1 = fmaf(mu, s1, fmaf(alpha, x2 - x0, fmaf(beta, c3, 0.1f * x1))); \
    float n2 = fmaf(mu, s2, fmaf(alpha, x3 - x1, fmaf(beta, c4, 0.1f * x2))); \
    float n3 = fmaf(mu, s3, fmaf(alpha, x4 - x2, fmaf(beta, c0, 0.1f * x3))); \
    float n4 = fmaf(mu, s4, fmaf(alpha, x0 - x3, fmaf(beta, c1, 0.1f * x4))); \
    x0 = __builtin_amdgcn_fractf(n0);                                         \
    x1 = __builtin_amdgcn_fractf(n1);                                         \
    x2 = __builtin_amdgcn_fractf(n2);                                         \
    x3 = __builtin_amdgcn_fractf(n3);                                         \
    x4 = __builtin_amdgcn_fractf(n4);                                         \
  } while (0)

// Kernel B: one chaotic chain per lane (B=64 -> 2 wave32 waves on separate
// WGPs). Inherently serial (65736 dependent steps); latency/trans-issue bound.
// Writes chaos[b][p]: per-thread sequential stores (fill cache lines over 16
// consecutive steps), 16 MB total -> negligible vs the ~1 ms serial compute.
__global__ __launch_bounds__(32) void chaos_gen_kernel(
    const float* __restrict__ key, float* __restrict__ chaos, int B) {
  int b = blockIdx.x * 32 + threadIdx.x;
  if (b >= B) return;
  float k1 = key[2 * b + 0];
  float k2 = key[2 * b + 1];

  // Accurate libm sin/cos for the one-time init (large arguments).
  float x0 = fmodf(fabsf(sinf(k1 * 10.f + k2 * 1.f)) + 0.1f, 0.9f);
  float x1 = fmodf(fabsf(sinf(k1 * 20.f + k2 * 2.f)) + 0.1f, 0.9f);
  float x2 = fmodf(fabsf(sinf(k1 * 30.f + k2 * 3.f)) + 0.1f, 0.9f);
  float x3 = fmodf(fabsf(sinf(k1 * 40.f + k2 * 5.f)) + 0.1f, 0.9f);
  float x4 = fmodf(fabsf(sinf(k1 * 50.f + k2 * 7.f)) + 0.1f, 0.9f);
  float mu    = 0.99f + fmodf(fabsf(sinf(k1 * 100.f)), 0.01f);
  float alpha = 0.40f + fmodf(fabsf(cosf(k2 * 100.f)), 0.10f);
  float beta  = 0.15f + fmodf(fabsf(sinf(k1 * 200.f + k2 * 200.f)), 0.10f);

  float* __restrict__ dst = chaos + (size_t)b * HW_TOT;

  for (int i = 0; i < WARMUP; ++i) {
    CHAOS_STEP();
  }
  for (int p = 0; p < HW_TOT; ++p) {
    CHAOS_STEP();
    dst[p] = __builtin_amdgcn_fractf(x0 + x1 + x2 + x3 + x4);
  }
}

extern "C" void kernel_launch(void* const* d_in, const int* in_sizes, int n_in,
                              void* d_out, int out_size, void* d_ws, size_t ws_size,
                              hipStream_t stream) {
  const float* images   = (const float*)d_in[0];  // (64,3,256,256) fp32
  const float* key      = (const float*)d_in[1];  // (64,2) fp32
  const int*   iters    = (const int*)d_in[2];    // scalar int
  const float* strength = (const float*)d_in[3];  // (1,) fp32
  float* out   = (float*)d_out;
  float* chaos = (float*)d_ws;                    // B * 65536 fp32 = 16 MB

  const int B = in_sizes[1] / 2;                  // 64

  chaos_gen_kernel<<<(B + 31) / 32, 32, 0, stream>>>(key, chaos, B);

  dim3 grid(HW_TOT / TILE, B);
  cipher_apply_kernel<<<grid, 256, 0, stream>>>(images, chaos, iters, strength, out);
}